// PropModule_4973572128834
// MI455X (gfx1250) — compile-verified
//
#include <hip/hip_runtime.h>

#define B_    8
#define N_    1024
#define R_    8192
#define DIN_  32
#define NF_   128
#define DOUT_ 3

typedef float v2f __attribute__((ext_vector_type(2)));
typedef float v8f __attribute__((ext_vector_type(8)));

__device__ __forceinline__ v8f wmma_f32(v2f a, v2f b, v8f c) {
  // D = A(16x4,f32) * B(4x16,f32) + C(16x16,f32)
  return __builtin_amdgcn_wmma_f32_16x16x4_f32(false, a, false, b, (short)0, c,
                                               false, false);
}

__device__ __forceinline__ v8f v8zero() {
  v8f v;
#pragma unroll
  for (int i = 0; i < 8; ++i) v[i] = 0.0f;
  return v;
}

// B-fragment for a 4x16 tile of a row-major [K,128] weight matrix.
// Lane layout: lanes 0-15 hold K = k0,k0+1 ; lanes 16-31 hold K = k0+2,k0+3.
// kk = k0 + 2*half already folded in by caller.
__device__ __forceinline__ v2f ldw(const float* __restrict__ W, int kk, int n) {
  v2f r;
  r.x = W[(size_t)kk * NF_ + n];
  r.y = W[(size_t)(kk + 1) * NF_ + n];
  return r;
}

// ---------------------------------------------------------------------------
// Collapse dense one-hot Rr/Rs [B,N,R] into integer index arrays [B*R].
// ---------------------------------------------------------------------------
__global__ __launch_bounds__(256) void k_indices(const float* __restrict__ Rr,
                                                 const float* __restrict__ Rs,
                                                 int* __restrict__ recv,
                                                 int* __restrict__ send) {
  int t = blockIdx.x * blockDim.x + threadIdx.x;  // 0 .. B*R-1
  int b = t / R_;
  int r = t - b * R_;
  const float* cr = Rr + (size_t)b * N_ * R_ + r;
  const float* cs = Rs + (size_t)b * N_ * R_ + r;
  int ir = 0, is = 0;
  for (int n = 0; n < N_; ++n) {
    float vr = *cr;
    float vs = *cs;
    ir = (vr > 0.5f) ? n : ir;
    is = (vs > 0.5f) ? n : is;
    cr += R_;
    cs += R_;
  }
  recv[t] = ir;
  send[t] = is;
}

__global__ __launch_bounds__(256) void k_zero(float* __restrict__ p, int n) {
  for (int i = blockIdx.x * blockDim.x + threadIdx.x; i < n;
       i += gridDim.x * blockDim.x)
    p[i] = 0.0f;
}

// ---------------------------------------------------------------------------
// Particle encoder: relu(relu(state @ w0 + b0) @ w1 + b1)   [B*N,32] -> [B*N,128]
// One wave per 16-row tile; layer-1 input re-staged through per-wave LDS.
// ---------------------------------------------------------------------------
__global__ __launch_bounds__(128) void k_penc(
    const float* __restrict__ state, const float* __restrict__ w0,
    const float* __restrict__ b0, const float* __restrict__ w1,
    const float* __restrict__ b1, float* __restrict__ out) {
  __shared__ __align__(16) float hb[4][16][132];
  const int lane = threadIdx.x & 31;
  const int w = threadIdx.x >> 5;
  const int lr = lane & 15;
  const int hf = lane >> 4;
  const int m0 = (blockIdx.x * 4 + w) * 16;

  const float* arow = state + (size_t)(m0 + lr) * DIN_;
  v8f acc[8];
#pragma unroll
  for (int nt = 0; nt < 8; ++nt) acc[nt] = v8zero();

  // layer 0: K = 32
#pragma unroll
  for (int ks = 0; ks < 8; ++ks) {
    const int kk = ks * 4 + hf * 2;
    v2f a;
    a.x = arow[kk];
    a.y = arow[kk + 1];
#pragma unroll
    for (int nt = 0; nt < 8; ++nt)
      acc[nt] = wmma_f32(a, ldw(w0, kk, nt * 16 + lr), acc[nt]);
  }
#pragma unroll
  for (int nt = 0; nt < 8; ++nt) {
    const float bia = b0[nt * 16 + lr];
#pragma unroll
    for (int i = 0; i < 8; ++i) {
      float v = acc[nt][i] + bia;
      hb[w][i + hf * 8][nt * 16 + lr] = fmaxf(v, 0.0f);
    }
    acc[nt] = v8zero();
  }

  // layer 1: K = 128 from LDS
#pragma unroll 2
  for (int ks = 0; ks < 32; ++ks) {
    const int kk = ks * 4 + hf * 2;
    v2f a = *(const v2f*)&hb[w][lr][kk];
#pragma unroll
    for (int nt = 0; nt < 8; ++nt)
      acc[nt] = wmma_f32(a, ldw(w1, kk, nt * 16 + lr), acc[nt]);
  }
#pragma unroll
  for (int nt = 0; nt < 8; ++nt) {
    const float bia = b1[nt * 16 + lr];
#pragma unroll
    for (int i = 0; i < 8; ++i) {
      float v = acc[nt][i] + bia;
      out[(size_t)(m0 + i + hf * 8) * NF_ + nt * 16 + lr] = fmaxf(v, 0.0f);
    }
  }
}

// ---------------------------------------------------------------------------
// Relation encoder: 3-layer MLP over [state[recv] | state[send] | Ra] (K=65).
// K=64 via WMMA + rank-1 update for the Ra column.
// ---------------------------------------------------------------------------
__global__ __launch_bounds__(128) void k_renc(
    const float* __restrict__ state, const int* __restrict__ recv,
    const int* __restrict__ send, const float* __restrict__ Ra,
    const float* __restrict__ w0, const float* __restrict__ b0,
    const float* __restrict__ w1, const float* __restrict__ b1,
    const float* __restrict__ w2, const float* __restrict__ b2,
    float* __restrict__ out) {
  __shared__ __align__(16) float hb[4][16][132];
  const int lane = threadIdx.x & 31;
  const int w = threadIdx.x >> 5;
  const int lr = lane & 15;
  const int hf = lane >> 4;
  const int m0 = (blockIdx.x * 4 + w) * 16;  // flat relation row base
  const int bb = m0 >> 13;                   // / R_
  const int row = m0 + lr;

  const float* rrow = state + ((size_t)bb * N_ + recv[row]) * DIN_;
  const float* srow = state + ((size_t)bb * N_ + send[row]) * DIN_;

  v8f acc[8];
#pragma unroll
  for (int nt = 0; nt < 8; ++nt) acc[nt] = v8zero();

  // layer 0: K = 64 (state_r | state_s)
#pragma unroll
  for (int ks = 0; ks < 16; ++ks) {
    const int kk = ks * 4 + hf * 2;
    const float* src = (kk < DIN_) ? (rrow + kk) : (srow + (kk - DIN_));
    v2f a;
    a.x = src[0];
    a.y = src[1];
#pragma unroll
    for (int nt = 0; nt < 8; ++nt)
      acc[nt] = wmma_f32(a, ldw(w0, kk, nt * 16 + lr), acc[nt]);
  }
  // rank-1 Ra contribution (weight row 64), bias, relu -> LDS
  float rav[8];
#pragma unroll
  for (int i = 0; i < 8; ++i) rav[i] = Ra[m0 + i + hf * 8];
#pragma unroll
  for (int nt = 0; nt < 8; ++nt) {
    const float w64 = w0[64 * NF_ + nt * 16 + lr];
    const float bia = b0[nt * 16 + lr];
#pragma unroll
    for (int i = 0; i < 8; ++i) {
      float v = acc[nt][i] + rav[i] * w64 + bia;
      hb[w][i + hf * 8][nt * 16 + lr] = fmaxf(v, 0.0f);
    }
    acc[nt] = v8zero();
  }

  // layer 1: K = 128
#pragma unroll 2
  for (int ks = 0; ks < 32; ++ks) {
    const int kk = ks * 4 + hf * 2;
    v2f a = *(const v2f*)&hb[w][lr][kk];
#pragma unroll
    for (int nt = 0; nt < 8; ++nt)
      acc[nt] = wmma_f32(a, ldw(w1, kk, nt * 16 + lr), acc[nt]);
  }
#pragma unroll
  for (int nt = 0; nt < 8; ++nt) {
    const float bia = b1[nt * 16 + lr];
#pragma unroll
    for (int i = 0; i < 8; ++i) {
      float v = acc[nt][i] + bia;
      hb[w][i + hf * 8][nt * 16 + lr] = fmaxf(v, 0.0f);
    }
    acc[nt] = v8zero();
  }

  // layer 2: K = 128
#pragma unroll 2
  for (int ks = 0; ks < 32; ++ks) {
    const int kk = ks * 4 + hf * 2;
    v2f a = *(const v2f*)&hb[w][lr][kk];
#pragma unroll
    for (int nt = 0; nt < 8; ++nt)
      acc[nt] = wmma_f32(a, ldw(w2, kk, nt * 16 + lr), acc[nt]);
  }
#pragma unroll
  for (int nt = 0; nt < 8; ++nt) {
    const float bia = b2[nt * 16 + lr];
#pragma unroll
    for (int i = 0; i < 8; ++i) {
      float v = acc[nt][i] + bia;
      out[(size_t)(m0 + i + hf * 8) * NF_ + nt * 16 + lr] = fmaxf(v, 0.0f);
    }
  }
}

// ---------------------------------------------------------------------------
// Relation propagator (per pstep iteration):
//   relation_effect = relu([renc | peff[recv] | peff[send]] @ rp_w + rp_b)
//   eagg[recv] += relation_effect   (atomic scatter-add)
// rp_w (384x128, 192KB) staged through LDS in 4 transposed 96-row chunks.
// ---------------------------------------------------------------------------
#define KC 96
__global__ __launch_bounds__(128) void k_relprop(
    const float* __restrict__ renc, const float* __restrict__ peff,
    const int* __restrict__ recv, const int* __restrict__ send,
    const float* __restrict__ rpw, const float* __restrict__ rpb,
    float* __restrict__ eagg) {
  __shared__ __align__(16) float wl[128][KC + 2];  // [n][k_local], stride 98
  const int tid = threadIdx.x;
  const int lane = tid & 31;
  const int w = tid >> 5;
  const int lr = lane & 15;
  const int hf = lane >> 4;
  const int m0 = (blockIdx.x * 4 + w) * 16;
  const int bb = m0 >> 13;
  const int row = m0 + lr;

  const float* ra = renc + (size_t)row * NF_;
  const float* rr = peff + ((size_t)bb * N_ + recv[row]) * NF_;
  const float* rs = peff + ((size_t)bb * N_ + send[row]) * NF_;

  int tgt[8];
#pragma unroll
  for (int i = 0; i < 8; ++i)
    tgt[i] = (bb * N_ + recv[m0 + i + hf * 8]) * NF_;

  v8f acc[8];
#pragma unroll
  for (int nt = 0; nt < 8; ++nt) acc[nt] = v8zero();

  for (int c = 0; c < 4; ++c) {
    // cooperative transposed chunk load: rpw rows [c*96, c*96+96)
    for (int idx = tid; idx < KC * 32; idx += 128) {
      const int k = idx >> 5;
      const int n4 = (idx & 31) * 4;
      float4 vv = *(const float4*)&rpw[(size_t)(c * KC + k) * NF_ + n4];
      wl[n4 + 0][k] = vv.x;
      wl[n4 + 1][k] = vv.y;
      wl[n4 + 2][k] = vv.z;
      wl[n4 + 3][k] = vv.w;
    }
    __syncthreads();

#pragma unroll 2
    for (int ks = 0; ks < KC / 4; ++ks) {
      const int kl = ks * 4 + hf * 2;  // chunk-local k
      const int kk = c * KC + kl;      // global k in [0,384)
      const float* src =
          (kk < NF_) ? (ra + kk)
                     : ((kk < 2 * NF_) ? (rr + kk - NF_) : (rs + kk - 2 * NF_));
      v2f a;
      a.x = src[0];
      a.y = src[1];
#pragma unroll
      for (int nt = 0; nt < 8; ++nt) {
        v2f bf = *(const v2f*)&wl[nt * 16 + lr][kl];
        acc[nt] = wmma_f32(a, bf, acc[nt]);
      }
    }
    __syncthreads();
  }

#pragma unroll
  for (int nt = 0; nt < 8; ++nt) {
    const float bia = rpb[nt * 16 + lr];
#pragma unroll
    for (int i = 0; i < 8; ++i) {
      float v = fmaxf(acc[nt][i] + bia, 0.0f);
      atomicAdd(&eagg[tgt[i] + nt * 16 + lr], v);
    }
  }
}

// ---------------------------------------------------------------------------
// Particle propagator: peff = relu([penc | eagg] @ pp_w + pp_b + peff)
// ---------------------------------------------------------------------------
__global__ __launch_bounds__(128) void k_partprop(
    const float* __restrict__ penc, const float* __restrict__ eagg,
    const float* __restrict__ ppw, const float* __restrict__ ppb,
    float* __restrict__ peff) {
  const int lane = threadIdx.x & 31;
  const int w = threadIdx.x >> 5;
  const int lr = lane & 15;
  const int hf = lane >> 4;
  const int m0 = (blockIdx.x * 4 + w) * 16;

  const float* r1 = penc + (size_t)(m0 + lr) * NF_;
  const float* r2 = eagg + (size_t)(m0 + lr) * NF_;

  v8f acc[8];
#pragma unroll
  for (int nt = 0; nt < 8; ++nt) acc[nt] = v8zero();

#pragma unroll 2
  for (int ks = 0; ks < 64; ++ks) {  // K = 256
    const int kk = ks * 4 + hf * 2;
    const float* src = (kk < NF_) ? (r1 + kk) : (r2 + kk - NF_);
    v2f a;
    a.x = src[0];
    a.y = src[1];
#pragma unroll
    for (int nt = 0; nt < 8; ++nt)
      acc[nt] = wmma_f32(a, ldw(ppw, kk, nt * 16 + lr), acc[nt]);
  }
#pragma unroll
  for (int nt = 0; nt < 8; ++nt) {
    const float bia = ppb[nt * 16 + lr];
#pragma unroll
    for (int i = 0; i < 8; ++i) {
      const size_t idx = (size_t)(m0 + i + hf * 8) * NF_ + nt * 16 + lr;
      float v = acc[nt][i] + bia + peff[idx];
      peff[idx] = fmaxf(v, 0.0f);
    }
  }
}

// ---------------------------------------------------------------------------
// Predictor: out = relu(peff @ pr_w0 + pr_b0) @ pr_w1 + pr_b1   (D_OUT = 3)
// ---------------------------------------------------------------------------
__global__ __launch_bounds__(128) void k_pred(
    const float* __restrict__ peff, const float* __restrict__ w0,
    const float* __restrict__ b0, const float* __restrict__ w1,
    const float* __restrict__ b1, float* __restrict__ out) {
  __shared__ __align__(16) float hb[4][16][132];
  const int lane = threadIdx.x & 31;
  const int w = threadIdx.x >> 5;
  const int lr = lane & 15;
  const int hf = lane >> 4;
  const int m0 = (blockIdx.x * 4 + w) * 16;

  const float* arow = peff + (size_t)(m0 + lr) * NF_;
  v8f acc[8];
#pragma unroll
  for (int nt = 0; nt < 8; ++nt) acc[nt] = v8zero();

#pragma unroll 2
  for (int ks = 0; ks < 32; ++ks) {
    const int kk = ks * 4 + hf * 2;
    v2f a;
    a.x = arow[kk];
    a.y = arow[kk + 1];
#pragma unroll
    for (int nt = 0; nt < 8; ++nt)
      acc[nt] = wmma_f32(a, ldw(w0, kk, nt * 16 + lr), acc[nt]);
  }
#pragma unroll
  for (int nt = 0; nt < 8; ++nt) {
    const float bia = b0[nt * 16 + lr];
#pragma unroll
    for (int i = 0; i < 8; ++i) {
      float v = acc[nt][i] + bia;
      hb[w][i + hf * 8][nt * 16 + lr] = fmaxf(v, 0.0f);
    }
  }

  // tiny final layer (128 -> 3), scalar dot products
  for (int j = lane; j < 16 * DOUT_; j += 32) {
    const int rr = j / DOUT_;
    const int cc = j - rr * DOUT_;
    float s = b1[cc];
    for (int k = 0; k < NF_; ++k) s += hb[w][rr][k] * w1[k * DOUT_ + cc];
    out[(size_t)(m0 + rr) * DOUT_ + cc] = s;
  }
}

// ---------------------------------------------------------------------------
extern "C" void kernel_launch(void* const* d_in, const int* in_sizes, int n_in,
                              void* d_out, int out_size, void* d_ws,
                              size_t ws_size, hipStream_t stream) {
  const float* state = (const float*)d_in[0];
  const float* Rr = (const float*)d_in[1];
  const float* Rs = (const float*)d_in[2];
  const float* Ra = (const float*)d_in[3];
  // d_in[4] = pstep (traced constant 3)
  const float* pe_w0 = (const float*)d_in[5];
  const float* pe_b0 = (const float*)d_in[6];
  const float* pe_w1 = (const float*)d_in[7];
  const float* pe_b1 = (const float*)d_in[8];
  const float* re_w0 = (const float*)d_in[9];
  const float* re_b0 = (const float*)d_in[10];
  const float* re_w1 = (const float*)d_in[11];
  const float* re_b1 = (const float*)d_in[12];
  const float* re_w2 = (const float*)d_in[13];
  const float* re_b2 = (const float*)d_in[14];
  const float* rp_w = (const float*)d_in[15];
  const float* rp_b = (const float*)d_in[16];
  const float* pp_w = (const float*)d_in[17];
  const float* pp_b = (const float*)d_in[18];
  const float* pr_w0 = (const float*)d_in[19];
  const float* pr_b0 = (const float*)d_in[20];
  const float* pr_w1 = (const float*)d_in[21];
  const float* pr_b1 = (const float*)d_in[22];
  float* out = (float*)d_out;

  char* ws = (char*)d_ws;
  int* recv = (int*)ws;                               // 256 KB
  int* send = (int*)(ws + (size_t)B_ * R_ * 4);       // 256 KB
  float* penc = (float*)(ws + (size_t)B_ * R_ * 8);   // 4 MB
  float* peff = penc + (size_t)B_ * N_ * NF_;         // 4 MB
  float* eagg = peff + (size_t)B_ * N_ * NF_;         // 4 MB
  float* renc = eagg + (size_t)B_ * N_ * NF_;         // 32 MB

  k_indices<<<(B_ * R_) / 256, 256, 0, stream>>>(Rr, Rs, recv, send);
  k_zero<<<256, 256, 0, stream>>>(peff, B_ * N_ * NF_);
  k_penc<<<(B_ * N_ / 16) / 4, 128, 0, stream>>>(state, pe_w0, pe_b0, pe_w1,
                                                 pe_b1, penc);
  k_renc<<<(B_ * R_ / 16) / 4, 128, 0, stream>>>(state, recv, send, Ra, re_w0,
                                                 re_b0, re_w1, re_b1, re_w2,
                                                 re_b2, renc);
  for (int it = 0; it < 3; ++it) {
    k_zero<<<256, 256, 0, stream>>>(eagg, B_ * N_ * NF_);
    k_relprop<<<(B_ * R_ / 16) / 4, 128, 0, stream>>>(renc, peff, recv, send,
                                                      rp_w, rp_b, eagg);
    k_partprop<<<(B_ * N_ / 16) / 4, 128, 0, stream>>>(penc, eagg, pp_w, pp_b,
                                                       peff);
  }
  k_pred<<<(B_ * N_ / 16) / 4, 128, 0, stream>>>(peff, pr_w0, pr_b0, pr_w1,
                                                 pr_b1, out);
}